// sparsemax_21122649162197
// MI455X (gfx1250) — compile-verified
//
#include <hip/hip_runtime.h>
#include <stdint.h>

#define SPX_DIM   512
#define SPX_WAVES 8
#define SPX_BLOCK (SPX_WAVES * 32)
#define SPX_RPW   2   // rows per wave: double-buffered TDM pipeline

typedef unsigned int u32x4 __attribute__((ext_vector_type(4)));
typedef int          i32x4 __attribute__((ext_vector_type(4)));
typedef int          i32x8 __attribute__((ext_vector_type(8)));

__device__ __forceinline__ float spx_wave_fsum(float v) {
#pragma unroll
  for (int m = 16; m >= 1; m >>= 1) v += __shfl_xor(v, m, 32);
  return v;
}
__device__ __forceinline__ int spx_wave_isum(int v) {
#pragma unroll
  for (int m = 16; m >= 1; m >>= 1) v += __shfl_xor(v, m, 32);
  return v;
}

#if __has_builtin(__builtin_amdgcn_tensor_load_to_lds)
// Issue one TDM descriptor: 1-D tile, 512 x f32 (2KB), global -> LDS.
__device__ __forceinline__ void spx_tdm_issue(const float* src, uint32_t lds) {
  const uint64_t ga = (uint64_t)(uintptr_t)src;
  u32x4 g0;
  g0.x = 1u;                                                  // count=1 (valid), user mode
  g0.y = lds;                                                 // lds_addr (byte offset)
  g0.z = (uint32_t)ga;                                        // global_addr[31:0]
  g0.w = (uint32_t)((ga >> 32) & 0x01FFFFFFu) | 0x80000000u;  // global_addr[56:32] | type=2

  i32x8 g1;
  g1[0] = 0x00020000;                      // workgroup_mask=0, data_size=2 (4 bytes)
  g1[1] = (int)((uint32_t)SPX_DIM << 16);  // tensor_dim0[15:0]=512 in bits[63:48]
  g1[2] = 0x00010000;                      // tensor_dim1=1
  g1[3] = (int)((uint32_t)SPX_DIM << 16);  // tile_dim0=512 in bits[127:112]
  g1[4] = 0;                               // tile_dim1=0, tile_dim2=0 (1-D tile)
  g1[5] = SPX_DIM;                         // tensor_dim0_stride
  g1[6] = 0;
  g1[7] = 0;

  i32x4 gz = {0, 0, 0, 0};
#if defined(__clang_major__) && (__clang_major__ >= 23)
  i32x8 gz8 = {0, 0, 0, 0, 0, 0, 0, 0};
  __builtin_amdgcn_tensor_load_to_lds(g0, g1, gz, gz, gz8, 0);
#else
  __builtin_amdgcn_tensor_load_to_lds(g0, g1, gz, gz, 0);
#endif
}
#endif

// Load one staged row (LDS), solve the sparsemax threshold exactly (Michelot),
// and write p = max(0, z - tau) with fully coalesced float4 stores.
__device__ __forceinline__ void spx_row_compute(const float (&srow)[SPX_DIM],
                                                float* __restrict__ dst, int lane) {
  float z[16];
  float s = 0.0f;
#pragma unroll
  for (int j = 0; j < 4; ++j) {
    const float4 v = *(const float4*)(&srow[j * 128 + lane * 4]);
    z[4 * j + 0] = v.x; z[4 * j + 1] = v.y; z[4 * j + 2] = v.z; z[4 * j + 3] = v.w;
    s += (v.x + v.y) + (v.z + v.w);
  }

  // Mean-center (output is shift-invariant; matches reference numerics)
  s = spx_wave_fsum(s);
  const float mean = s * (1.0f / (float)SPX_DIM);
#pragma unroll
  for (int i = 0; i < 16; ++i) z[i] -= mean;

  // Michelot: tau <- (sum_{z>tau} z - 1)/|{z>tau}|; support only shrinks,
  // equal count => identical support => tau exact. Wave-uniform break.
  float tau = -__builtin_inff();
  int k_prev = -1;
#pragma unroll 1
  for (int it = 0; it < 64; ++it) {
    float ssum = 0.0f;
    int cnt = 0;
#pragma unroll
    for (int i = 0; i < 16; ++i) {
      if (z[i] > tau) { ssum += z[i]; cnt++; }
    }
    ssum = spx_wave_fsum(ssum);
    cnt  = spx_wave_isum(cnt);
    tau = (ssum - 1.0f) / (float)cnt;
    if (cnt == k_prev) break;
    k_prev = cnt;
  }

#pragma unroll
  for (int j = 0; j < 4; ++j) {
    float4 o;
    o.x = fmaxf(0.0f, z[4 * j + 0] - tau);
    o.y = fmaxf(0.0f, z[4 * j + 1] - tau);
    o.z = fmaxf(0.0f, z[4 * j + 2] - tau);
    o.w = fmaxf(0.0f, z[4 * j + 3] - tau);
    *(float4*)(dst + j * 128 + lane * 4) = o;
  }
}

__global__ __launch_bounds__(SPX_BLOCK) void sparsemax_tdm_kernel(
    const float* __restrict__ x, float* __restrict__ out, int batch) {
  __shared__ float smem[SPX_WAVES][SPX_RPW][SPX_DIM];

  const int lane = threadIdx.x & 31;
  const int wave = threadIdx.x >> 5;
  const int row0 = (blockIdx.x * SPX_WAVES + wave) * SPX_RPW;
  if (row0 >= batch) return;            // wave-uniform
  const bool has1 = (row0 + 1) < batch; // wave-uniform

#if __has_builtin(__builtin_amdgcn_tensor_load_to_lds)
  // Pipeline: issue both DMAs, then overlap row-1 DMA with row-0 compute.
  spx_tdm_issue(x + (size_t)row0 * SPX_DIM,
                (uint32_t)(uintptr_t)(&smem[wave][0][0]));
  if (has1) {
    spx_tdm_issue(x + (size_t)(row0 + 1) * SPX_DIM,
                  (uint32_t)(uintptr_t)(&smem[wave][1][0]));
    __builtin_amdgcn_s_wait_tensorcnt(1);   // first DMA complete (in-order)
    asm volatile("" ::: "memory");
    spx_row_compute(smem[wave][0], out + (size_t)row0 * SPX_DIM, lane);
    __builtin_amdgcn_s_wait_tensorcnt(0);   // second DMA complete
    asm volatile("" ::: "memory");
    spx_row_compute(smem[wave][1], out + (size_t)(row0 + 1) * SPX_DIM, lane);
  } else {
    __builtin_amdgcn_s_wait_tensorcnt(0);
    asm volatile("" ::: "memory");
    spx_row_compute(smem[wave][0], out + (size_t)row0 * SPX_DIM, lane);
  }
#else
  // Fallback: stage rows into LDS with plain coalesced loads, then compute.
#pragma unroll
  for (int r = 0; r < SPX_RPW; ++r) {
    if (r == 1 && !has1) break;
    const float* src = x + (size_t)(row0 + r) * SPX_DIM;
#pragma unroll
    for (int j = 0; j < 4; ++j)
      *(float4*)(&smem[wave][r][j * 128 + lane * 4]) =
          *(const float4*)(src + j * 128 + lane * 4);
    spx_row_compute(smem[wave][r], out + (size_t)(row0 + r) * SPX_DIM, lane);
  }
#endif
}

extern "C" void kernel_launch(void* const* d_in, const int* in_sizes, int n_in,
                              void* d_out, int out_size, void* d_ws, size_t ws_size,
                              hipStream_t stream) {
  const float* x = (const float*)d_in[0];
  float* out = (float*)d_out;
  const int batch = in_sizes[0] / SPX_DIM;  // 65536
  const int rows_per_block = SPX_WAVES * SPX_RPW;
  const int blocks = (batch + rows_per_block - 1) / rows_per_block;
  sparsemax_tdm_kernel<<<blocks, SPX_BLOCK, 0, stream>>>(x, out, batch);
}